// AEDecoder_55671366090777
// MI455X (gfx1250) — compile-verified
//
#include <hip/hip_runtime.h>

typedef float v2f __attribute__((ext_vector_type(2)));
typedef float v8f __attribute__((ext_vector_type(8)));

#define NGROUP 20000
#define WDIM 8
#define KDIM 16
#define TDIM 1500
#define BDIM 256
#define NWAVE 16
#define NTHREAD (NWAVE * 32)

// Branchless SELU: scale*max(x,0) + scale*alpha*(exp(min(x,0)) - 1)
// (at x>0 the exp term is exp(0)-1 = 0; at x<=0 the max term is 0)
__device__ __forceinline__ float selu_f(float x) {
    const float scale       = 1.0507009873554805f;
    const float scale_alpha = 1.0507009873554805f * 1.6732632423543772f;
    float pos = fmaxf(x, 0.0f);
    float neg = __expf(fminf(x, 0.0f)) - 1.0f;
    return scale * pos + scale_alpha * neg;
}

__global__ __launch_bounds__(NTHREAD)
void aedecoder_kernel(const float* __restrict__ features,
                      const int*   __restrict__ tf_idx,
                      const float* __restrict__ W1,
                      const float* __restrict__ b1,
                      const float* __restrict__ Wm,
                      const float* __restrict__ bm,
                      const float* __restrict__ Wf,
                      const float* __restrict__ bf,
                      float* __restrict__ out)
{
    // All weight/bias slabs zero-padded to N=16 so fragment loads need no predication.
    __shared__ float s_w1[16][KDIM];          // [n][k]; n>=8 rows are 0
    __shared__ float s_wm[2][16][WDIM];       // [i][n][w]; n>=8 rows are 0
    __shared__ float s_b1[16];                // n>=8 -> 0
    __shared__ float s_bm[2][16];             // n>=8 -> 0
    __shared__ float s_wf[WDIM];
    __shared__ float s_bf[1];
    __shared__ int   s_idx[KDIM];
    __shared__ float s_h[NWAVE][16][17];      // per-wave 16x16 tile, stride 17 pad

    const int g = blockIdx.x;
    const int t = threadIdx.x;

    // ---- cooperative stage of per-group params into LDS (zero-padded) ----
    if (t < 256) {                                       // s_w1: 16x16
        int n = t >> 4, k = t & 15;
        s_w1[n][k] = (n < WDIM) ? W1[g * (WDIM * KDIM) + n * KDIM + k] : 0.0f;
    } else {                                             // s_wm: 2x16x8
        int j = t - 256;                                 // 0..255
        int i = j >> 7, r = j & 127;
        int n = r >> 3, w = r & 7;
        s_wm[i][n][w] = (n < WDIM) ? Wm[(i * NGROUP + g) * (WDIM * WDIM) + n * WDIM + w] : 0.0f;
    }
    if (t < KDIM) s_idx[t] = tf_idx[g * KDIM + t];
    if (t >= 448 && t < 464) {
        int n = t - 448;
        s_b1[n] = (n < WDIM) ? b1[g * WDIM + n] : 0.0f;
    } else if (t < 496) {
        if (t >= 464) {
            int j = t - 464; int i = j >> 4; int n = j & 15;
            s_bm[i][n] = (n < WDIM) ? bm[(i * NGROUP + g) * WDIM + n] : 0.0f;
        }
    } else if (t < 504) {
        s_wf[t - 496] = Wf[g * WDIM + (t - 496)];
    } else if (t == 504) {
        s_bf[0] = bf[g];
    }
    __syncthreads();

    const int wave = t >> 5;
    const int lane = t & 31;
    const int half = lane >> 4;     // high K bit of A/B fragments
    const int m    = lane & 15;     // A role: row; B/C role: column n
    const int b0   = wave * 16;     // batch-row tile base

    // ================= layer 1: [16x16] x [16x16(zero-padded)] =================
    v8f c;
    {
        float bias = s_b1[m];                    // 0 for n>=8
        #pragma unroll
        for (int r = 0; r < 8; ++r) c[r] = bias; // bias folded into accumulator
    }
    #pragma unroll
    for (int kk = 0; kk < 4; ++kk) {
        const int k0 = kk * 4 + half * 2;        // this lane supplies K = k0, k0+1
        v2f a, b;
        {
            const int row = b0 + m;              // gather xg from L2-resident features
            a.x = features[row * TDIM + s_idx[k0 + 0]];
            a.y = features[row * TDIM + s_idx[k0 + 1]];
        }
        b.x = s_w1[m][k0 + 0];                   // B[k][n] = W1[n][k] (rows n>=8 are 0)
        b.y = s_w1[m][k0 + 1];
        c = __builtin_amdgcn_wmma_f32_16x16x4_f32(false, a, false, b, (short)0, c, false, false);
    }
    #pragma unroll
    for (int r = 0; r < 8; ++r) c[r] = selu_f(c[r]);

    // D layout: VGPR r holds (row r+8*half, col m); cols 8..15 are exact zeros.
    #pragma unroll
    for (int r = 0; r < 8; ++r) s_h[wave][r + 8 * half][m] = c[r];
    __builtin_amdgcn_wave_barrier();

    // ================= layers 2..3: [16x8] x [8x16(zero-padded)] =================
    #pragma unroll
    for (int layer = 0; layer < 2; ++layer) {
        float bias = s_bm[layer][m];
        #pragma unroll
        for (int r = 0; r < 8; ++r) c[r] = bias;
        #pragma unroll
        for (int kk = 0; kk < 2; ++kk) {
            const int k0 = kk * 4 + half * 2;    // K in [0,8)
            v2f a, b;
            a.x = s_h[wave][m][k0 + 0];          // A from LDS-transposed h
            a.y = s_h[wave][m][k0 + 1];
            b.x = s_wm[layer][m][k0 + 0];        // B[w][n] = Wm[n][w] (rows n>=8 are 0)
            b.y = s_wm[layer][m][k0 + 1];
            c = __builtin_amdgcn_wmma_f32_16x16x4_f32(false, a, false, b, (short)0, c, false, false);
        }
        #pragma unroll
        for (int r = 0; r < 8; ++r) c[r] = selu_f(c[r]);
        __builtin_amdgcn_wave_barrier();         // A-reads precede overwrite (same-wave DS order)
        #pragma unroll
        for (int r = 0; r < 8; ++r) s_h[wave][r + 8 * half][m] = c[r];
        __builtin_amdgcn_wave_barrier();
    }

    // ================= final: out[b] = h . Wf + bf =================
    if (lane < 16) {
        float acc = s_bf[0];
        #pragma unroll
        for (int w = 0; w < WDIM; ++w) acc += s_h[wave][lane][w] * s_wf[w];
        out[(b0 + lane) * NGROUP + g] = acc;     // column write; lines merge in 192MB L2
    }
}

extern "C" void kernel_launch(void* const* d_in, const int* in_sizes, int n_in,
                              void* d_out, int out_size, void* d_ws, size_t ws_size,
                              hipStream_t stream) {
    const float* features = (const float*)d_in[0];
    const int*   tf_idx   = (const int*)  d_in[1];
    const float* W1       = (const float*)d_in[2];
    const float* b1       = (const float*)d_in[3];
    const float* Wm       = (const float*)d_in[4];
    const float* bm       = (const float*)d_in[5];
    const float* Wf       = (const float*)d_in[6];
    const float* bf       = (const float*)d_in[7];
    float* out = (float*)d_out;

    aedecoder_kernel<<<NGROUP, NTHREAD, 0, stream>>>(
        features, tf_idx, W1, b1, Wm, bm, Wf, bf, out);
}